// BornMayerHugginsForce_20933670601273
// MI455X (gfx1250) — compile-verified
//
#include <hip/hip_runtime.h>
#include <hip/hip_bf16.h>

// ---------------------------------------------------------------------------
// Born-Mayer-Huggins pair potential over an edge list (gfx1250 / MI455X).
//
// - stream edge indices coalesced (only real HBM traffic; pos/types/forces
//   are L2-resident at 1.2MB / 0.4MB / 1.2MB vs 192MB L2)
// - early-out on squared cutoff: ~96.6% of edges culled before gathers/exp/atomics
// - 5x 4x4 parameter tables in LDS via CDNA5 async global->LDS copy
// - force scatter via global_atomic_add_f32; wave32 shuffle reduce for E/virial
// ---------------------------------------------------------------------------

#if defined(__gfx1250__) && __has_builtin(__builtin_amdgcn_global_load_async_to_lds_b32)
#define BMH_HAVE_ASYNC_LDS 1
#else
#define BMH_HAVE_ASYNC_LDS 0
#endif

typedef __attribute__((address_space(1))) int* bmh_gptr_t;
typedef __attribute__((address_space(3))) int* bmh_lptr_t;

__device__ __forceinline__ void bmh_wait_asynccnt0() {
#if defined(__gfx1250__)
#if __has_builtin(__builtin_amdgcn_s_wait_asynccnt)
  __builtin_amdgcn_s_wait_asynccnt(0);
#else
  asm volatile("s_wait_asynccnt 0" ::: "memory");
#endif
#endif
}

__global__ __launch_bounds__(256) void bmh_edge_kernel(
    const float* __restrict__ pos,   // [N,3]
    const float* __restrict__ Atab,  // [4,4]
    const float* __restrict__ Ctab,
    const float* __restrict__ Dtab,
    const float* __restrict__ Rtab,  // rho
    const float* __restrict__ Stab,  // sig
    const int* __restrict__ ei,      // [E]
    const int* __restrict__ ej,      // [E]
    const int* __restrict__ atype,   // [N]
    const int* __restrict__ cut_p,   // scalar
    const int* __restrict__ bl_p,    // scalar
    int nE, int nAtoms,
    float* __restrict__ out)         // [0]=energy, [1..3N]=forces, [1+3N]=virial
{
  // --- 5 x 16 parameter LUT in LDS -------------------------------------
  __shared__ float lut[80];  // [0:16)=A [16:32)=C [32:48)=D [48:64)=rho [64:80)=sig
  {
    const int t = threadIdx.x;
    if (t < 80) {
      const float* g = (t < 16) ? Atab : (t < 32) ? Ctab : (t < 48) ? Dtab
                       : (t < 64) ? Rtab : Stab;
      g += (t & 15);
#if BMH_HAVE_ASYNC_LDS
      __builtin_amdgcn_global_load_async_to_lds_b32(
          (bmh_gptr_t)(float*)g,
          (bmh_lptr_t)&lut[t],
          /*offset=*/0, /*cpol=*/0);
#else
      lut[t] = __ldg(g);
#endif
    }
#if BMH_HAVE_ASYNC_LDS
    bmh_wait_asynccnt0();
#endif
    __syncthreads();
  }

  const float rc  = (float)(*cut_p);
  const float bl  = (float)(*bl_p);
  const float rc2 = rc * rc;
  const float inv_bl = 1.0f / bl;

  float* __restrict__ F = out + 1;      // forces base
  const int virial_idx  = 1 + 3 * nAtoms;

  float esum = 0.0f;
  float vsum = 0.0f;

  const int stride = (int)(gridDim.x * blockDim.x);
  for (int e = (int)(blockIdx.x * blockDim.x + threadIdx.x); e < nE; e += stride) {
    // prefetch the next edge indices this lane will touch (global_prefetch_b8)
    if (e + stride < nE) {
      __builtin_prefetch(&ei[e + stride], 0, 0);
      __builtin_prefetch(&ej[e + stride], 0, 0);
    }

    const int i = ei[e];
    const int j = ej[e];

    // minimum-image displacement (L2-resident gathers)
    const float* pi = pos + 3 * i;
    const float* pj = pos + 3 * j;
    float dx = pi[0] - pj[0];
    float dy = pi[1] - pj[1];
    float dz = pi[2] - pj[2];
    dx -= bl * rintf(dx * inv_bl);
    dy -= bl * rintf(dy * inv_bl);
    dz -= bl * rintf(dz * inv_bl);

    const float r2 = fmaf(dx, dx, fmaf(dy, dy, dz * dz));

    // ~96.6% of edges die here: skip type gathers, exp, and all atomics.
    if (r2 < rc2) {
      float r = sqrtf(r2);
      r = fmaxf(r, 1e-6f);
      const float inv_r  = 1.0f / r;
      const float inv_r2 = inv_r * inv_r;
      const float inv_r6 = inv_r2 * inv_r2 * inv_r2;
      const float inv_r8 = inv_r6 * inv_r2;

      const int k   = (atype[i] << 2) + atype[j];
      const float Ae  = lut[k];
      const float Ce  = lut[16 + k];
      const float De  = lut[32 + k];
      const float rhe = lut[48 + k];
      const float sge = lut[64 + k];

      const float inv_rho = 1.0f / rhe;
      const float expo    = expf((sge - r) * inv_rho);

      const float U = Ae * expo - Ce * inv_r6 + De * inv_r8;
      const float dU_dr = -Ae * inv_rho * expo
                          + 6.0f * Ce * inv_r6 * inv_r
                          - 8.0f * De * inv_r8 * inv_r;
      const float Fmag = -dU_dr;

      const float s  = Fmag * inv_r;
      const float fx = s * dx;
      const float fy = s * dy;
      const float fz = s * dz;

      atomicAdd(&F[3 * i + 0],  fx);
      atomicAdd(&F[3 * i + 1],  fy);
      atomicAdd(&F[3 * i + 2],  fz);
      atomicAdd(&F[3 * j + 0], -fx);
      atomicAdd(&F[3 * j + 1], -fy);
      atomicAdd(&F[3 * j + 2], -fz);

      esum += 0.5f * U;
      vsum += Fmag * r;
    }
  }

  // wave32 tree reduction, one atomic per wave
  for (int off = 16; off > 0; off >>= 1) {
    esum += __shfl_down(esum, off, 32);
    vsum += __shfl_down(vsum, off, 32);
  }
  if ((threadIdx.x & 31) == 0) {
    if (esum != 0.0f) atomicAdd(&out[0], esum);
    if (vsum != 0.0f) atomicAdd(&out[virial_idx], vsum);
  }
}

extern "C" void kernel_launch(void* const* d_in, const int* in_sizes, int n_in,
                              void* d_out, int out_size, void* d_ws, size_t ws_size,
                              hipStream_t stream) {
  (void)n_in; (void)d_ws; (void)ws_size;

  const float* pos  = (const float*)d_in[0];
  const float* Atab = (const float*)d_in[1];
  const float* Ctab = (const float*)d_in[2];
  const float* Dtab = (const float*)d_in[3];
  const float* Rtab = (const float*)d_in[4];
  const float* Stab = (const float*)d_in[5];
  const int*   eidx = (const int*)d_in[6];   // [2,E] flattened: rows contiguous
  const int*   atyp = (const int*)d_in[7];
  const int*   cutp = (const int*)d_in[8];
  const int*   blp  = (const int*)d_in[9];

  const int nE = in_sizes[6] / 2;
  const int nA = in_sizes[0] / 3;

  // energy + forces + virial, accumulated atomically -> zero first (capture-safe)
  (void)hipMemsetAsync(d_out, 0, (size_t)out_size * sizeof(float), stream);

  const int threads = 256;
  int blocks = (nE + threads - 1) / threads;
  if (blocks > 4096) blocks = 4096;

  bmh_edge_kernel<<<blocks, threads, 0, stream>>>(
      pos, Atab, Ctab, Dtab, Rtab, Stab,
      eidx, eidx + nE, atyp, cutp, blp,
      nE, nA, (float*)d_out);
}